// Ising_model_54348516163822
// MI455X (gfx1250) — compile-verified
//
#include <hip/hip_runtime.h>

typedef float v2f __attribute__((ext_vector_type(2)));
typedef float v8f __attribute__((ext_vector_type(8)));

#define NN 8192
#define CSPLIT 16                  // column segments per 16-row panel
#define SEGCOLS (NN / CSPLIT)      // 512
#define NPANEL (NN / 16)           // 512
#define NWAVES (NPANEL * CSPLIT)   // 8192
#define WAVES_PER_BLOCK 8

// Accumulate WMMA dual-reduction over column range [kbeg, kend).
// Precondition: kbeg/kend are wave-uniform (SGPR) and (kend-kbeg) % 16 == 0.
// A tile: 16 rows x 4 cols of M (one nontemporal float2 per lane).
// B: col0 = s[k..k+3] (quad path), col1 = ones (sum path), rest zero.
__device__ __forceinline__ void panel_span(const float* __restrict__ pA,
                                           const float* __restrict__ pS,
                                           int kbeg, int kend,
                                           bool sel0, float cs1, v8f& acc)
{
    for (int k0 = kbeg; k0 < kend; k0 += 16) {
#pragma unroll
        for (int u = 0; u < 4; ++u) {
            const int k = k0 + 4 * u;
            v2f a  = __builtin_nontemporal_load((const v2f*)(pA + k));
            v2f sv = *(const v2f*)(pS + k);
            v2f b;
            b.x = sel0 ? sv.x : cs1;           // B vgpr0: K = 2*hi
            b.y = sel0 ? sv.y : cs1;           // B vgpr1: K = 2*hi + 1
            acc = __builtin_amdgcn_wmma_f32_16x16x4_f32(
                /*neg_a=*/false, a, /*neg_b=*/false, b,
                /*c_mod=*/(short)0, acc, /*reuse_a=*/false, /*reuse_b=*/false);
        }
    }
}

__global__ __launch_bounds__(256) void ising_panel_kernel(
    const float* __restrict__ M, const float* __restrict__ s,
    float* __restrict__ partial)
{
    const int lane = threadIdx.x & 31;
    // Force wave-uniform values into SGPRs so loop control is scalar and
    // EXEC stays all-ones around the WMMAs.
    const int wv   = __builtin_amdgcn_readfirstlane((int)(threadIdx.x >> 5));
    const int wid  = blockIdx.x * WAVES_PER_BLOCK + wv;
    const int panel = wid / CSPLIT;
    const int seg   = wid % CSPLIT;
    const int r0    = panel * 16;
    const int c0    = seg * SEGCOLS;
    const int c1    = c0 + SEGCOLS;

    const int   laneN = lane & 15;
    const int   hi    = lane >> 4;                  // which half-wave
    const bool  sel0  = (laneN == 0);               // D column 0: quad path
    const float cs1   = (laneN == 1) ? 1.0f : 0.0f; // D column 1: sum path

    const float* pA = M + (size_t)(r0 + laneN) * NN + (hi << 1);
    const float* pS = s + (hi << 1);

    v8f cup = {0.f,0.f,0.f,0.f,0.f,0.f,0.f,0.f};    // upper (quad+sum)
    v8f clo = {0.f,0.f,0.f,0.f,0.f,0.f,0.f,0.f};    // lower (sum only)

    // Both ranges have length % 16 == 0 (c0,r0 multiples of 16).
    const int loEnd = (c1 < r0) ? c1 : r0;            // lower: [c0, loEnd)
    const int upBeg = (c0 > r0 + 16) ? c0 : r0 + 16;  // upper: [upBeg, c1)

    if (c0 < loEnd) panel_span(pA, pS, c0, loEnd, sel0, cs1, clo);
    if (upBeg < c1) panel_span(pA, pS, upBeg, c1, sel0, cs1, cup);

    // 16x16 diagonal block: scalar (256 elems across 32 lanes)
    float dsum = 0.f, dquad = 0.f, dlin = 0.f;
    if (r0 >= c0 && r0 < c1) {
#pragma unroll
        for (int t = 0; t < 8; ++t) {
            int e  = lane + 32 * t;
            int rr = e >> 4, cc = e & 15;
            float m  = M[(size_t)(r0 + rr) * NN + (r0 + cc)];
            float sr = s[r0 + rr], sc = s[r0 + cc];
            dsum += m;
            if (cc > rr)  dquad += m * sr * sc;
            if (cc == rr) dlin  += m * sr;
        }
    }

    // Fold D accumulators. D layout: lane = N; vgpr v -> row v (lanes 0-15)
    // or row v+8 (lanes 16-31). Col 0 = s-weighted sums, col 1 = plain sums.
    float qp = 0.f, sp = 0.f;
    if (laneN == 0) {
#pragma unroll
        for (int v = 0; v < 8; ++v)
            qp += cup[v] * s[r0 + (hi ? 8 : 0) + v];
    } else if (laneN == 1) {
#pragma unroll
        for (int v = 0; v < 8; ++v)
            sp += cup[v] + clo[v];
    }

    float r = 0.25f * (sp + dsum) - 0.5f * (qp + dquad) - 0.5f * dlin;

#pragma unroll
    for (int off = 16; off > 0; off >>= 1)
        r += __shfl_xor(r, off, 32);

    if (lane == 0) partial[wid] = r;
}

__global__ __launch_bounds__(256) void ising_final_reduce(
    const float* __restrict__ partial, float* __restrict__ out)
{
    __shared__ float sm[256];
    float a = 0.f;
    for (int i = threadIdx.x; i < NWAVES; i += 256) a += partial[i];
    sm[threadIdx.x] = a;
    __syncthreads();
    for (int st = 128; st > 0; st >>= 1) {
        if ((int)threadIdx.x < st) sm[threadIdx.x] += sm[threadIdx.x + st];
        __syncthreads();
    }
    if (threadIdx.x == 0) out[0] = sm[0];
}

extern "C" void kernel_launch(void* const* d_in, const int* in_sizes, int n_in,
                              void* d_out, int out_size, void* d_ws, size_t ws_size,
                              hipStream_t stream) {
    const float* M = (const float*)d_in[0];   // info_mtx [8192*8192] f32
    const float* s = (const float*)d_in[1];   // state    [8192]      f32
    float* partial = (float*)d_ws;            // NWAVES floats (32 KB)

    ising_panel_kernel<<<NWAVES / WAVES_PER_BLOCK, 256, 0, stream>>>(M, s, partial);
    ising_final_reduce<<<1, 256, 0, stream>>>(partial, (float*)d_out);
}